// SPOPFNSAttention_16140487098657
// MI455X (gfx1250) — compile-verified
//
#include <hip/hip_runtime.h>
#include <math.h>

typedef __attribute__((ext_vector_type(2))) float v2f;
typedef __attribute__((ext_vector_type(8))) float v8f;

#define NN 2048
#define DD 64
#define HH 8
#define BHN 16              // B*H
#define MASK_VALF 10000.0f
#define EPSF 1e-7f

// score = (1 + g)^-5, g = acos(clip(qk)) masked to 10000 when key is padded
__device__ __forceinline__ float score_of(float qk, int keep) {
    float q = fminf(fmaxf(qk, -1.0f + EPSF), 1.0f - EPSF);
    float g = acosf(q);              // R = 1.0
    g = keep ? g : MASK_VALF;
    float t = 1.0f + g;              // bandwidth = 1.0
    float inv = 1.0f / t;
    float i2 = inv * inv;
    return i2 * i2 * inv;            // t^-5  (= t^(-d_int-alpha))
}

// ---------------------------------------------------------------------------
// Kernel 1: per-(b,h) column degrees  ncinv[bh, m] = rsqrt(sum_n score[n,m])
// One wave owns a 16-column strip; streams all 128 row tiles with fp32 WMMA.
// ---------------------------------------------------------------------------
__global__ __launch_bounds__(128) void spop_colsum_kernel(
    const float* __restrict__ Q, const float* __restrict__ K,
    const int* __restrict__ mask, float* __restrict__ ncinv) {
    const int lane = threadIdx.x & 31;
    const int wave = threadIdx.x >> 5;
    const int gw   = blockIdx.x * 4 + wave;      // global wave id
    const int bh   = gw >> 7;                    // / (N/16)
    const int cm   = (gw & 127) << 4;            // column base
    const int b    = bh >> 3;                    // / H
    const size_t base = (size_t)bh * NN * DD;

    const int ln = lane & 15;                    // N (and M) index inside tile
    const int lh = lane >> 4;                    // half-wave selector

    // B fragments: K rows cm+ln, contiguous float2 at k = 4*kc + 2*lh
    v2f kfr[16];
#pragma unroll
    for (int kc = 0; kc < 16; ++kc)
        kfr[kc] = *(const v2f*)(K + base + (size_t)(cm + ln) * DD + kc * 4 + 2 * lh);

    const int keep = mask[b * NN + cm + ln];
    float colAcc = 0.0f;

    for (int rt = 0; rt < 128; ++rt) {
        const int rn = rt << 4;
        v2f afr[16];
#pragma unroll
        for (int kc = 0; kc < 16; ++kc)
            afr[kc] = *(const v2f*)(Q + base + (size_t)(rn + ln) * DD + kc * 4 + 2 * lh);
        v8f c = {};
#pragma unroll
        for (int kc = 0; kc < 16; ++kc)
            c = __builtin_amdgcn_wmma_f32_16x16x4_f32(
                    false, afr[kc], false, kfr[kc], (short)0, c, false, false);
#pragma unroll
        for (int r = 0; r < 8; ++r)
            colAcc += score_of(c[r], keep);
    }
    // column n=ln lives in lanes ln and ln+16: combine halves
    colAcc += __shfl_xor(colAcc, 16, 32);
    if (lane < 16)
        ncinv[bh * NN + cm + lane] = rsqrtf(colAcc);   // n_c^(-a), a = 0.5
}

// ---------------------------------------------------------------------------
// Kernel 2: single-pass output. One wave owns 16 rows; streams column tiles,
// accumulates O += (score * nc^-1/2) @ V with fp32 WMMA, plus n_r and S.
// ---------------------------------------------------------------------------
__global__ __launch_bounds__(128) void spop_attn_out_kernel(
    const float* __restrict__ Q, const float* __restrict__ K,
    const float* __restrict__ V, const int* __restrict__ mask,
    const float* __restrict__ ncinv, float* __restrict__ out) {
    __shared__ float lds[4][16 * 17];            // per-wave padded P tile
    const int lane = threadIdx.x & 31;
    const int wave = threadIdx.x >> 5;
    const int gw   = blockIdx.x * 4 + wave;
    const int bh   = gw >> 7;
    const int rn   = (gw & 127) << 4;            // row base
    const int b    = bh >> 3;
    const size_t base = (size_t)bh * NN * DD;
    const int ln = lane & 15;
    const int lh = lane >> 4;
    float* myLds = lds[wave];

    // hoisted A fragments of Q (rows rn..rn+15)
    v2f qfr[16];
#pragma unroll
    for (int kc = 0; kc < 16; ++kc)
        qfr[kc] = *(const v2f*)(Q + base + (size_t)(rn + ln) * DD + kc * 4 + 2 * lh);

    v8f oacc[4] = {};                            // O tile: 16 rows x 64 cols
    float rAcc[8], sAcc[8];
#pragma unroll
    for (int r = 0; r < 8; ++r) { rAcc[r] = 0.0f; sAcc[r] = 0.0f; }

    for (int ct = 0; ct < 128; ++ct) {
        const int cm = ct << 4;
        v2f kfr[16];
#pragma unroll
        for (int kc = 0; kc < 16; ++kc)
            kfr[kc] = *(const v2f*)(K + base + (size_t)(cm + ln) * DD + kc * 4 + 2 * lh);
        const int keep = mask[b * NN + cm + ln];
        const float ncv = ncinv[bh * NN + cm + ln];

        v8f c = {};
#pragma unroll
        for (int kc = 0; kc < 16; ++kc)
            c = __builtin_amdgcn_wmma_f32_16x16x4_f32(
                    false, qfr[kc], false, kfr[kc], (short)0, c, false, false);

        // scores -> P = score * nc^-1/2; stage to LDS for layout transpose
#pragma unroll
        for (int r = 0; r < 8; ++r) {
            float s = score_of(c[r], keep);
            rAcc[r] += s;                        // n_r partial (this column)
            float p = s * ncv;
            sAcc[r] += p;                        // S partial
            myLds[(r + 8 * lh) * 17 + ln] = p;   // P[M][N], M=r+8*lh, N=ln
        }
        // O += P(16x16) x V(16x64): 4 D-groups x 4 k-chunks of fp32 WMMA
#pragma unroll
        for (int dg = 0; dg < 4; ++dg) {
#pragma unroll
            for (int kc = 0; kc < 4; ++kc) {
                v2f a, bv;
                a.x = myLds[ln * 17 + kc * 4 + 2 * lh];       // A: M=ln, K pair
                a.y = myLds[ln * 17 + kc * 4 + 2 * lh + 1];
                const float* vp = V + base
                                + (size_t)(cm + kc * 4 + 2 * lh) * DD
                                + dg * 16 + ln;               // B: K row pair, N=ln
                bv.x = vp[0];
                bv.y = vp[DD];
                oacc[dg] = __builtin_amdgcn_wmma_f32_16x16x4_f32(
                               false, a, false, bv, (short)0, oacc[dg], false, false);
            }
        }
    }

    // reduce row sums across the 16 lanes of each half-wave
#pragma unroll
    for (int off = 1; off < 16; off <<= 1) {
#pragma unroll
        for (int r = 0; r < 8; ++r) {
            rAcc[r] += __shfl_xor(rAcc[r], off, 32);
            sAcc[r] += __shfl_xor(sAcc[r], off, 32);
        }
    }
    // out_row = n_r^-a * Oacc_row / max(n_r^-a * S_row, 1e-12)
    float scl[8];
#pragma unroll
    for (int r = 0; r < 8; ++r) {
        float rs = rsqrtf(rAcc[r]);
        float denom = fmaxf(rs * sAcc[r], 1e-12f);
        scl[r] = rs / denom;
    }
#pragma unroll
    for (int dg = 0; dg < 4; ++dg)
#pragma unroll
        for (int r = 0; r < 8; ++r)
            out[(size_t)(bh * NN + rn + r + 8 * lh) * DD + dg * 16 + ln] =
                oacc[dg][r] * scl[r];
}

extern "C" void kernel_launch(void* const* d_in, const int* in_sizes, int n_in,
                              void* d_out, int out_size, void* d_ws, size_t ws_size,
                              hipStream_t stream) {
    (void)in_sizes; (void)n_in; (void)out_size; (void)ws_size;
    const float* Q    = (const float*)d_in[0];
    const float* K    = (const float*)d_in[1];
    const float* V    = (const float*)d_in[2];
    const int*   mask = (const int*)d_in[3];
    float* out   = (float*)d_out;
    float* ncinv = (float*)d_ws;                 // BH*N floats = 128 KB

    // 16 heads * 128 strips of 16 / 4 waves per block = 512 blocks, 128 thr
    dim3 block(128), grid(512);
    spop_colsum_kernel<<<grid, block, 0, stream>>>(Q, K, mask, ncinv);
    spop_attn_out_kernel<<<grid, block, 0, stream>>>(Q, K, V, mask, ncinv, out);
}